// Uncondition9RotL_24799141167110
// MI455X (gfx1250) — compile-verified
//
#include <hip/hip_runtime.h>

// CDNA5 / gfx1250: wave32, WMMA 16x16x4 f32, async global<->LDS staging.
typedef __attribute__((ext_vector_type(2))) float v2f;
typedef __attribute__((ext_vector_type(8))) float v8f;
typedef int v4i __attribute__((vector_size(16)));   // b128 payload type for async LDS ops

#define TILE 256   // matrices per block (8 waves * 32)

#if __has_builtin(__builtin_amdgcn_global_load_async_to_lds_b128)
#define HAVE_ASYNC_LD 1
#else
#define HAVE_ASYNC_LD 0
#endif
#if __has_builtin(__builtin_amdgcn_global_store_async_from_lds_b128)
#define HAVE_ASYNC_ST 1
#else
#define HAVE_ASYNC_ST 0
#endif

typedef __attribute__((address_space(1))) v4i* gv4p;  // global (AS1) int4*
typedef __attribute__((address_space(3))) v4i* lv4p;  // LDS (AS3) int4*

__global__ __launch_bounds__(TILE) void polar3x3_wmma_kernel(
    const float* __restrict__ rot,     // [nmat, 3, 3]
    const float* __restrict__ mat,     // [3, 3] shared
    float* __restrict__ out,           // [nmat, 3, 3]
    float* __restrict__ logdet,        // [nmat]
    int nmat)
{
    __shared__ float lds_in[TILE * 9];   // staged rot tiles
    __shared__ float lds_nm[TILE * 9];   // new_mat = mat @ rot, then results

    const int t         = threadIdx.x;
    const int lane      = t & 31;
    const int wavebase  = t & ~31;               // first local matrix of this wave
    const int blockBase = blockIdx.x * TILE;
    const int valid     = min(TILE, nmat - blockBase);
    const float* gin    = rot + (size_t)blockBase * 9;

    // ---------------- Phase 1: coalesced global -> LDS staging (async b128) ----
    if (valid == TILE) {
        const float4* g4 = (const float4*)gin;
#if HAVE_ASYNC_LD
        #pragma unroll
        for (int i = t; i < TILE * 9 / 4; i += TILE) {
            __builtin_amdgcn_global_load_async_to_lds_b128(
                (gv4p)(g4 + i), (lv4p)(lds_in + i * 4), 0, 0);
        }
        asm volatile("s_wait_asynccnt 0x0" ::: "memory");
#else
        float4* l4 = (float4*)lds_in;
        #pragma unroll
        for (int i = t; i < TILE * 9 / 4; i += TILE) l4[i] = g4[i];
#endif
    } else {
        int nflt = valid * 9;
        for (int i = t; i < nflt; i += TILE) lds_in[i] = gin[i];
    }
    __syncthreads();

    // ---------------- Phase 2: new_mat = mat @ rot via V_WMMA_F32_16X16X4_F32 --
    // Per wave: 32 matrices -> 96 (matrix,column) pairs -> 6 WMMAs of 16 rows.
    // A(16x4):  A[m][k] = rot[n(m)][k][c(m)]  (K=3 padded with 0)
    // B(4x16):  B[k][j] = mat[j][k] for j<3,k<3 else 0   (i.e. mat^T)
    // D(16x16): D[m][j] = sum_k rot[n][k][c]*mat[j][k] = new_mat[n][j][c]
    {
        const bool lo = (lane < 16);
        const int  j  = lane & 15;               // D/B column held by this lane
        const int  jc = (j < 3) ? j : 0;
        const float mj0 = mat[jc * 3 + 0];
        const float mj1 = mat[jc * 3 + 1];
        const float mj2 = mat[jc * 3 + 2];
        v2f B;
        B.x = (j < 3) ? (lo ? mj0 : mj2) : 0.0f; // rows 0 / 2 of B
        B.y = (j < 3 && lo) ? mj1 : 0.0f;        // rows 1 / 3 (row 3 = pad)

        const int m = lane & 15;                 // A row held by this lane
        // Base LDS addresses for D extraction (compile-time immediate offsets).
        float* nm_lo = lds_nm + wavebase * 9 + lane * 3;        // lanes 0..2
        float* nm_hi = lds_nm + wavebase * 9 + (lane - 16) * 3; // lanes 16..18

        #pragma unroll
        for (int g = 0; g < 6; ++g) {
            const int p  = g * 16 + m;           // (matrix,column) pair index
            const int ln = p / 3;
            const int c  = p - ln * 3;
            const int n  = wavebase + ln;
            v2f A;
            A.x = lo ? lds_in[n * 9 + 0 + c] : lds_in[n * 9 + 6 + c]; // K=0 / K=2
            A.y = lo ? lds_in[n * 9 + 3 + c] : 0.0f;                  // K=1 / K=3(pad)
            v8f Cz = {};
            v8f D = __builtin_amdgcn_wmma_f32_16x16x4_f32(
                false, A, false, B, (short)0, Cz, false, false);

            union { v8f v; float f[8]; } u; u.v = D;
            // Lanes 0..2 hold D rows 0..7 of cols 0..2; lanes 16..18 rows 8..15.
            // pp constant per unrolled (g,v) => lnn*9+cc folds to an immediate.
            if (lane < 3) {
                #pragma unroll
                for (int v = 0; v < 8; ++v) {
                    const int pp  = g * 16 + v;
                    const int lnn = pp / 3;
                    const int cc  = pp - lnn * 3;
                    nm_lo[lnn * 9 + cc] = u.f[v];
                }
            } else if (lane >= 16 && lane < 19) {
                #pragma unroll
                for (int v = 0; v < 8; ++v) {
                    const int pp  = g * 16 + v + 8;
                    const int lnn = pp / 3;
                    const int cc  = pp - lnn * 3;
                    nm_hi[lnn * 9 + cc] = u.f[v];
                }
            }
        }
    }
    __syncthreads();

    // ---------------- Phase 3: per-lane Newton-Higham polar factor -------------
    // X <- 0.5*(g*X + (1/g)*X^-T), g = (||X^-1||_F / ||X||_F)^(1/2).
    // X^-T = cof(X)/det. Converges to U@Vh of the SVD (incl. det<0 case).
    {
        float x[9];
        #pragma unroll
        for (int k = 0; k < 9; ++k) x[k] = lds_nm[t * 9 + k];

        #pragma unroll
        for (int it = 0; it < 7; ++it) {
            const float C0 = x[4]*x[8] - x[5]*x[7];
            const float C1 = x[5]*x[6] - x[3]*x[8];
            const float C2 = x[3]*x[7] - x[4]*x[6];
            const float C3 = x[2]*x[7] - x[1]*x[8];
            const float C4 = x[0]*x[8] - x[2]*x[6];
            const float C5 = x[1]*x[6] - x[0]*x[7];
            const float C6 = x[1]*x[5] - x[2]*x[4];
            const float C7 = x[2]*x[3] - x[0]*x[5];
            const float C8 = x[0]*x[4] - x[1]*x[3];
            const float det  = x[0]*C0 + x[1]*C1 + x[2]*C2;
            const float rdet = __builtin_amdgcn_rcpf(det);

            const float sX = x[0]*x[0]+x[1]*x[1]+x[2]*x[2]+x[3]*x[3]+x[4]*x[4]
                           + x[5]*x[5]+x[6]*x[6]+x[7]*x[7]+x[8]*x[8];
            const float sC = C0*C0+C1*C1+C2*C2+C3*C3+C4*C4+C5*C5+C6*C6+C7*C7+C8*C8;
            // g = (||X^-1||F / ||X||F)^(1/2) = (sC*rdet^2/sX)^(1/4)
            const float ratio = sC * rdet * rdet * __builtin_amdgcn_rcpf(sX);
            const float gsc = __builtin_sqrtf(__builtin_sqrtf(ratio));
            const float ka = 0.5f * gsc;
            const float kb = 0.5f * rdet * __builtin_amdgcn_rcpf(gsc);

            x[0] = ka*x[0] + kb*C0;  x[1] = ka*x[1] + kb*C1;  x[2] = ka*x[2] + kb*C2;
            x[3] = ka*x[3] + kb*C3;  x[4] = ka*x[4] + kb*C4;  x[5] = ka*x[5] + kb*C5;
            x[6] = ka*x[6] + kb*C6;  x[7] = ka*x[7] + kb*C7;  x[8] = ka*x[8] + kb*C8;
        }

        #pragma unroll
        for (int k = 0; k < 9; ++k) lds_nm[t * 9 + k] = x[k];
    }
    __syncthreads();

    // ---------------- Phase 4: coalesced LDS -> global stores (async b128) -----
    if (valid == TILE) {
        float4* o4 = (float4*)(out + (size_t)blockBase * 9);
#if HAVE_ASYNC_ST
        #pragma unroll
        for (int i = t; i < TILE * 9 / 4; i += TILE) {
            __builtin_amdgcn_global_store_async_from_lds_b128(
                (gv4p)(o4 + i), (lv4p)(lds_nm + i * 4), 0, 0);
        }
        // completion covered by implicit wait-idle at s_endpgm
#else
        const float4* l4 = (const float4*)lds_nm;
        #pragma unroll
        for (int i = t; i < TILE * 9 / 4; i += TILE) o4[i] = l4[i];
#endif
    } else {
        float* o = out + (size_t)blockBase * 9;
        int nflt = valid * 9;
        for (int i = t; i < nflt; i += TILE) o[i] = lds_nm[i];
    }
    if (t < valid) logdet[blockBase + t] = 0.0f;
}

extern "C" void kernel_launch(void* const* d_in, const int* in_sizes, int n_in,
                              void* d_out, int out_size, void* d_ws, size_t ws_size,
                              hipStream_t stream) {
    (void)n_in; (void)out_size; (void)d_ws; (void)ws_size;
    const float* rot = (const float*)d_in[0];
    const float* mat = (const float*)d_in[1];
    float* out = (float*)d_out;
    const int nmat = in_sizes[0] / 9;
    float* logdet = out + (size_t)nmat * 9;
    const int blocks = (nmat + TILE - 1) / TILE;
    polar3x3_wmma_kernel<<<blocks, TILE, 0, stream>>>(rot, mat, out, logdet, nmat);
}